// Node_node_50869592655513
// MI455X (gfx1250) — compile-verified
//
#include <hip/hip_runtime.h>

#define NN  100000
#define NE  600000
#define DIM 128
#define HID 256

typedef float v2f __attribute__((ext_vector_type(2)));
typedef float v8f __attribute__((ext_vector_type(8)));

// ---------------------------------------------------------------------------
// Stage 1: per-edge message + scatter-add into node accumulator (L2-resident)
// ---------------------------------------------------------------------------
__global__ __launch_bounds__(256) void edge_scatter(
    const float* __restrict__ x, const float* __restrict__ ea,
    const int* __restrict__ src, const int* __restrict__ dst,
    float* __restrict__ acc) {
  int e = blockIdx.x * 2 + (threadIdx.x >> 7);   // 2 edges per 256-thread block
  int d = threadIdx.x & (DIM - 1);
  int s = src[e];
  int t = dst[e];
  float m = x[s * DIM + d] + x[t * DIM + d] + ea[(size_t)e * DIM + d];
  unsafeAtomicAdd(&acc[s * DIM + d], m);       // global_atomic_add_f32
  unsafeAtomicAdd(&acc[t * DIM + d], m);
}

// ---------------------------------------------------------------------------
// Stage 2: h = node_new + (1 + eps - degree) * node_rep   (in place on acc)
// ---------------------------------------------------------------------------
__global__ __launch_bounds__(256) void residual(
    float* __restrict__ acc, const float* __restrict__ x,
    const float* __restrict__ degree, const float* __restrict__ eps) {
  int idx = blockIdx.x * 256 + threadIdx.x;    // N*DIM = 12.8M < 2^31
  int n = idx >> 7;
  float coef = 1.0f + eps[0] - degree[n];
  acc[idx] = acc[idx] + coef * x[idx];
}

// ---------------------------------------------------------------------------
// FP32 WMMA GEMM: Y[N, LDY] = act(A[N,K]) * B^T, B row-major [cols, K].
// One wave -> 16x64 output tile (4 accumulators reuse one A fragment).
// Optional fused BN+ReLU applied to A elements on load (layer-2 input).
// ---------------------------------------------------------------------------
template <int K, int LDY, int COLTILES, bool BN_A>
__global__ __launch_bounds__(128) void gemm_wmma(
    const float* __restrict__ A, const float* __restrict__ B,
    float* __restrict__ Y,
    const float* __restrict__ scaleA, const float* __restrict__ shiftA) {
  const int lane = threadIdx.x & 31;
  const int wave = threadIdx.x >> 5;
  const int tile = blockIdx.x * 4 + wave;
  const int rowBase = (tile / COLTILES) * 16;
  const int colBase = (tile % COLTILES) * 64;
  const int m16  = lane & 15;
  const int koff = (lane >> 4) * 2;            // lanes 0-15: K=k,k+1; 16-31: k+2,k+3

  const float* aP = A + (rowBase + m16) * K + koff;
  const float* bP = B + (colBase + m16) * K + koff;

  v8f acc0 = {}, acc1 = {}, acc2 = {}, acc3 = {};
#pragma unroll 4
  for (int k = 0; k < K; k += 4) {
    v2f a = *(const v2f*)(aP + k);
    if constexpr (BN_A) {
      int c = k + koff;
      a.x = fmaxf(fmaf(a.x, scaleA[c],     shiftA[c]),     0.0f);
      a.y = fmaxf(fmaf(a.y, scaleA[c + 1], shiftA[c + 1]), 0.0f);
    }
    v2f b0 = *(const v2f*)(bP + k);
    v2f b1 = *(const v2f*)(bP + 16 * K + k);
    v2f b2 = *(const v2f*)(bP + 32 * K + k);
    v2f b3 = *(const v2f*)(bP + 48 * K + k);
    acc0 = __builtin_amdgcn_wmma_f32_16x16x4_f32(false, a, false, b0, (short)0, acc0, false, false);
    acc1 = __builtin_amdgcn_wmma_f32_16x16x4_f32(false, a, false, b1, (short)0, acc1, false, false);
    acc2 = __builtin_amdgcn_wmma_f32_16x16x4_f32(false, a, false, b2, (short)0, acc2, false, false);
    acc3 = __builtin_amdgcn_wmma_f32_16x16x4_f32(false, a, false, b3, (short)0, acc3, false, false);
  }
  // D layout: VGPR v -> row rowBase + v + 8*(lane>>4), col colBase + (lane&15)
  float* yP = Y + (rowBase + 8 * (lane >> 4)) * LDY + colBase + m16;
#pragma unroll
  for (int v = 0; v < 8; ++v) {
    yP[v * LDY + 0]  = acc0[v];
    yP[v * LDY + 16] = acc1[v];
    yP[v * LDY + 32] = acc2[v];
    yP[v * LDY + 48] = acc3[v];
  }
}

// ---------------------------------------------------------------------------
// Column statistics: per-column sum / sum-of-squares over the batch dim.
// blockDim.x == C (#columns); each block covers rowsPer rows.
// ---------------------------------------------------------------------------
__global__ void colstats(const float* __restrict__ Y, int C, int rowsPer,
                         float* __restrict__ gsum, float* __restrict__ gsq) {
  int c = threadIdx.x;
  int r0 = blockIdx.x * rowsPer;
  int r1 = r0 + rowsPer;
  if (r1 > NN) r1 = NN;
  float s = 0.0f, sq = 0.0f;
  for (int r = r0; r < r1; ++r) {
    float y = Y[(size_t)r * C + c];
    s += y;
    sq += y * y;
  }
  unsafeAtomicAdd(&gsum[c], s);
  unsafeAtomicAdd(&gsq[c], sq);
}

__global__ void finalize_stats(const float* __restrict__ gsum,
                               const float* __restrict__ gsq,
                               const float* __restrict__ gamma,
                               const float* __restrict__ beta,
                               float* __restrict__ scale,
                               float* __restrict__ shift) {
  int c = threadIdx.x;
  float mu  = gsum[c] * (1.0f / NN);
  float var = gsq[c] * (1.0f / NN) - mu * mu;
  float inv = rsqrtf(var + 1e-5f);
  float sc  = gamma[c] * inv;
  scale[c] = sc;
  shift[c] = beta[c] - mu * sc;
}

// Final in-place BN + ReLU on d_out (C = DIM columns)
__global__ __launch_bounds__(256) void bn_relu_apply(
    float* __restrict__ Y, const float* __restrict__ scale,
    const float* __restrict__ shift) {
  int idx = blockIdx.x * 256 + threadIdx.x;
  int c = idx & (DIM - 1);
  Y[idx] = fmaxf(fmaf(Y[idx], scale[c], shift[c]), 0.0f);
}

// ---------------------------------------------------------------------------
extern "C" void kernel_launch(void* const* d_in, const int* in_sizes, int n_in,
                              void* d_out, int out_size, void* d_ws, size_t ws_size,
                              hipStream_t stream) {
  const float* node_rep  = (const float*)d_in[0];
  const float* edge_attr = (const float*)d_in[1];
  const float* degree    = (const float*)d_in[2];
  const float* eps       = (const float*)d_in[3];
  const float* W1        = (const float*)d_in[4];   // [HID, DIM]
  const float* g1        = (const float*)d_in[5];
  const float* b1        = (const float*)d_in[6];
  const float* W2        = (const float*)d_in[7];   // [DIM, HID]
  const float* g2        = (const float*)d_in[8];
  const float* b2        = (const float*)d_in[9];
  const int*   src       = (const int*)d_in[10];
  const int*   dst       = (const int*)d_in[11];
  float* out = (float*)d_out;

  // Workspace layout (floats): h[N*DIM] | Y1[N*HID] | stats
  float* h  = (float*)d_ws;
  float* Y1 = h + (size_t)NN * DIM;
  float* st = Y1 + (size_t)NN * HID;
  float* sum1 = st;            float* sq1   = st + HID;
  float* scl1 = st + 2 * HID;  float* shf1  = st + 3 * HID;
  float* sum2 = st + 4 * HID;  float* sq2   = sum2 + DIM;
  float* scl2 = sq2 + DIM;     float* shf2  = scl2 + DIM;

  hipMemsetAsync(h, 0, (size_t)NN * DIM * sizeof(float), stream);
  hipMemsetAsync(st, 0, (4 * HID + 4 * DIM) * sizeof(float), stream);

  // 1) edge messages + scatter-add (atomics into L2-resident accumulator)
  edge_scatter<<<NE / 2, 256, 0, stream>>>(node_rep, edge_attr, src, dst, h);

  // 2) degree-scaled residual (in place)
  residual<<<(NN * DIM) / 256, 256, 0, stream>>>(h, node_rep, degree, eps);

  // 3) Y1 = h @ W1^T   (16-row x 64-col tiles; 6250*4 waves / 4 per block)
  gemm_wmma<DIM, HID, HID / 64, false>
      <<<(NN / 16) * (HID / 64) / 4, 128, 0, stream>>>(h, W1, Y1, nullptr, nullptr);

  // 4) BN1 statistics + scale/shift
  colstats<<<400, HID, 0, stream>>>(Y1, HID, 250, sum1, sq1);
  finalize_stats<<<1, HID, 0, stream>>>(sum1, sq1, g1, b1, scl1, shf1);

  // 5) out_raw = relu(bn(Y1)) @ W2^T   (BN+ReLU fused into A-load)
  gemm_wmma<HID, DIM, DIM / 64, true>
      <<<(NN / 16) * (DIM / 64) / 4, 128, 0, stream>>>(Y1, W2, out, scl1, shf1);

  // 6) BN2 statistics + in-place BN+ReLU epilogue on d_out
  colstats<<<400, DIM, 0, stream>>>(out, DIM, 250, sum2, sq2);
  finalize_stats<<<1, DIM, 0, stream>>>(sum2, sq2, g2, b2, scl2, shf2);
  bn_relu_apply<<<(NN * DIM) / 256, 256, 0, stream>>>(out, scl2, shf2);
}